// Generator3DLUT_zero_73057393705705
// MI455X (gfx1250) — compile-verified
//
#include <hip/hip_runtime.h>
#include <cstdint>

// ---------------- problem constants (fixed by the reference) ----------------
#define LUT_D   33
#define LUT_C   (LUT_D * LUT_D * LUT_D)      // 35937 elements per LUT channel
#define BATCH   4
#define HW      (2048 * 2048)                // 4194304 pixels per plane
#define NQUADS  (BATCH * HW / 4)             // 4194304 float4-quads total
#define QPP     (HW / 4)                     // 1048576 quads per batch image
#define SHBYTES (2u * LUT_C * 4u)            // 287496 B: LUT channels 0,1 in LDS

static_assert(QPP == (1 << 20), "quad-per-plane must be 2^20 for shift math");
static_assert(SHBYTES <= 320u * 1024u, "must fit WGP LDS");

typedef unsigned int v4u __attribute__((ext_vector_type(4)));
typedef int          v8i __attribute__((ext_vector_type(8)));
typedef int          v4i __attribute__((ext_vector_type(4)));

// 8-byte LUT corner pair at 4-byte alignment (r0, r0+1 are adjacent in memory)
struct __attribute__((packed, aligned(4))) pf2 { float lo, hi; };

// ---- TDM: DMA LUT channels 0,1 (contiguous 2*35937 floats) global -> LDS ----
// D# per CDNA5 ISA ch.8: group0 = {count/flags, lds_addr, global_addr, type},
// group1 = dims/strides. 2D tile: dim0 = 35937 (one channel), dim1 = 2,
// tensor_dim0_stride = 35937 -> fully contiguous copy. workgroup_mask = 0
// (not in a cluster). data_size code 2 (= 4 bytes).
// amdgpu-toolchain (clang-23) builtin form: 6 args
//   (uint32x4 g0, int32x8 g1, int32x4, int32x4, int32x8, i32 cpol)
__device__ __forceinline__ void tdm_load_lut01(const float* gsrc, uint32_t lds_off)
{
    uint64_t ga = (uint64_t)(uintptr_t)gsrc;
    v4u g0;
    g0[0] = 1u;                                            // count=1, user desc
    g0[1] = lds_off;                                       // LDS byte address
    g0[2] = (uint32_t)ga;                                  // global_addr[31:0]
    g0[3] = ((uint32_t)(ga >> 32) & 0x01FFFFFFu)           // global_addr[56:32]
          | (2u << 30);                                    // type = 2 ("image")
    v8i g1;
    g1[0] = (int)(2u << 16);                               // data_size = 4B
    g1[1] = (int)((uint32_t)LUT_C << 16);                  // tensor_dim0 lo16
    g1[2] = (int)(2u << 16);                               // dim0 hi=0, tensor_dim1=2
    g1[3] = (int)((uint32_t)LUT_C << 16);                  // dim1 hi=0, tile_dim0=35937
    g1[4] = 2;                                             // tile_dim1=2, tile_dim2=0
    g1[5] = LUT_C;                                         // tensor_dim0_stride lo32
    g1[6] = 0;                                             // stride hi / dim1_stride lo
    g1[7] = 0;
    v4i z4 = {0, 0, 0, 0};                                 // groups 2/3 unused (<=2D)
    v8i z8 = {0, 0, 0, 0, 0, 0, 0, 0};
    __builtin_amdgcn_tensor_load_to_lds(g0, g1, z4, z4, z8, 0);
    __builtin_amdgcn_s_wait_tensorcnt(0);                  // TENSORcnt -> 0
}

// ------------- factored trilinear tap: 7 fused lerps per channel -------------
template <typename P>
__device__ __forceinline__ float tri_tap(P base, int i00, int i01, int i10, int i11,
                                         float fr, float fg, float fb)
{
    float a0 = base[i00], a1 = base[i00 + 1];              // fuses to ds_load_2addr
    float b0 = base[i01], b1 = base[i01 + 1];
    float c0 = base[i10], c1 = base[i10 + 1];
    float d0 = base[i11], d1 = base[i11 + 1];
    float v00 = fmaf(fr, a1 - a0, a0);
    float v01 = fmaf(fr, b1 - b0, b0);
    float v10 = fmaf(fr, c1 - c0, c0);
    float v11 = fmaf(fr, d1 - d0, d0);
    float v0  = fmaf(fg, v01 - v00, v00);
    float v1  = fmaf(fg, v11 - v10, v10);
    return fmaf(fb, v1 - v0, v0);
}

__device__ __forceinline__ float tri_tap_g(const float* __restrict__ base,
                                           int i00, int i01, int i10, int i11,
                                           float fr, float fg, float fb)
{
    pf2 a = *(const pf2*)(base + i00);                     // 8B gather, 4B aligned
    pf2 b = *(const pf2*)(base + i01);
    pf2 c = *(const pf2*)(base + i10);
    pf2 d = *(const pf2*)(base + i11);
    float v00 = fmaf(fr, a.hi - a.lo, a.lo);
    float v01 = fmaf(fr, b.hi - b.lo, b.lo);
    float v10 = fmaf(fr, c.hi - c.lo, c.lo);
    float v11 = fmaf(fr, d.hi - d.lo, d.lo);
    float v0  = fmaf(fg, v01 - v00, v00);
    float v1  = fmaf(fg, v11 - v10, v10);
    return fmaf(fb, v1 - v0, v0);
}

extern "C" __global__ void __launch_bounds__(1024, 1)
lut3d_trilinear_kernel(const float* __restrict__ x,
                       const float* __restrict__ lut,
                       float* __restrict__ out)
{
    extern __shared__ float sL[];   // LUT channels 0,1: 2*35937 floats

    // Wave 0 issues the tensor DMA; ISA: low 32 bits of a generic LDS pointer
    // are the LDS byte offset (aperture occupies the high bits).
    if (threadIdx.x < 32) {
        uint32_t lds_off = (uint32_t)(uintptr_t)(void*)sL;
        tdm_load_lut01(lut, lds_off);
    }
    __syncthreads();                                       // LDS LUT now valid

    const float* __restrict__ lut2 = lut + 2 * LUT_C;      // channel 2 stays in L2
    const int stride = (int)(gridDim.x * blockDim.x);

    for (int q = (int)(blockIdx.x * blockDim.x + threadIdx.x); q < NQUADS; q += stride) {
        // prefetch next iteration's pixel quads (R,G,B planes)
        int qn = q + stride;
        if (qn < NQUADS) {
            const float* pn = x + (size_t)(qn >> 20) * (3 * (size_t)HW)
                                + (size_t)((qn & (QPP - 1)) << 2);
            __builtin_prefetch(pn,            0, 0);       // global_prefetch_b8
            __builtin_prefetch(pn + HW,       0, 0);
            __builtin_prefetch(pn + 2 * HW,   0, 0);
        }

        const size_t base = (size_t)(q >> 20) * (3 * (size_t)HW)
                          + (size_t)((q & (QPP - 1)) << 2);
        const float4 R = *(const float4*)(x + base);            // 128-bit streams
        const float4 G = *(const float4*)(x + base + HW);
        const float4 B = *(const float4*)(x + base + 2 * HW);

        const float rv[4] = {R.x, R.y, R.z, R.w};
        const float gv[4] = {G.x, G.y, G.z, G.w};
        const float bv[4] = {B.x, B.y, B.z, B.w};
        float o0[4], o1[4], o2[4];

#pragma unroll
        for (int j = 0; j < 4; ++j) {
            float rr = rv[j] * 32.0f;
            float gg = gv[j] * 32.0f;
            float bb = bv[j] * 32.0f;
            int r0 = (int)rr; r0 = r0 < 0 ? 0 : (r0 > 31 ? 31 : r0);
            int g0 = (int)gg; g0 = g0 < 0 ? 0 : (g0 > 31 ? 31 : g0);
            int b0 = (int)bb; b0 = b0 < 0 ? 0 : (b0 > 31 ? 31 : b0);
            float fr = rr - (float)r0;
            float fg = gg - (float)g0;
            float fb = bb - (float)b0;
            int i00 = (b0 * LUT_D + g0) * LUT_D + r0;           // corner cell base
            int i01 = i00 + LUT_D;                              // g+1
            int i10 = i00 + LUT_D * LUT_D;                      // b+1
            int i11 = i10 + LUT_D;                              // b+1, g+1
            o0[j] = tri_tap(sL,          i00, i01, i10, i11, fr, fg, fb); // LDS
            o1[j] = tri_tap(sL + LUT_C,  i00, i01, i10, i11, fr, fg, fb); // LDS
            o2[j] = tri_tap_g(lut2,      i00, i01, i10, i11, fr, fg, fb); // global
        }

        float* ob = out + base;
        *(float4*)(ob)          = make_float4(o0[0], o0[1], o0[2], o0[3]);
        *(float4*)(ob + HW)     = make_float4(o1[0], o1[1], o1[2], o1[3]);
        *(float4*)(ob + 2 * HW) = make_float4(o2[0], o2[1], o2[2], o2[3]);
    }
}

extern "C" void kernel_launch(void* const* d_in, const int* in_sizes, int n_in,
                              void* d_out, int out_size, void* d_ws, size_t ws_size,
                              hipStream_t stream) {
    (void)in_sizes; (void)n_in; (void)d_ws; (void)ws_size; (void)out_size;
    const float* x   = (const float*)d_in[0];   // (4,3,2048,2048) f32
    const float* lut = (const float*)d_in[1];   // (3,33,33,33) f32
    float* out = (float*)d_out;                 // (4,3,2048,2048) f32

    // allow 287.5 KB of dynamic LDS (one workgroup per WGP)
    (void)hipFuncSetAttribute((const void*)lut3d_trilinear_kernel,
                              hipFuncAttributeMaxDynamicSharedMemorySize, (int)SHBYTES);

    dim3 grid(1024), block(1024);               // 4 grid-stride iters per thread
    lut3d_trilinear_kernel<<<grid, block, SHBYTES, stream>>>(x, lut, out);
}